// PrewittLayer_60696477827496
// MI455X (gfx1250) — compile-verified
//
#include <hip/hip_runtime.h>

// ---------------------------------------------------------------------------
// Prewitt layer, MI455X (gfx1250).
// Memory-bound: 617 MB traffic -> ~26us floor at 23.3 TB/s. WMMA irrelevant
// (0.6 GFLOP total). CDNA5 path: async global->LDS b128 staging, one thread
// per image row (full row in registers, edge cases resolved at compile time),
// b128 non-temporal streaming stores (write-once outputs, 411 MB > 192MB L2).
// ---------------------------------------------------------------------------

#define AS1 __attribute__((address_space(1)))
#define AS3 __attribute__((address_space(3)))

typedef int   v4i __attribute__((ext_vector_type(4)));
typedef float v4f __attribute__((ext_vector_type(4)));

constexpr int W              = 28;
constexpr int IMG_PIX        = 28 * 28;                   // 784
constexpr int IMGS_PER_BLOCK = 8;
constexpr int THREADS        = IMGS_PER_BLOCK * W;        // 224 = 7 waves, 1 row/thread
constexpr int PIX_PER_BLOCK  = IMGS_PER_BLOCK * IMG_PIX;  // 6272 floats = 25088 B LDS
constexpr int VEC4_PER_BLOCK = PIX_PER_BLOCK / 4;         // 1568 = 7 * 224 exactly

__global__ __launch_bounds__(THREADS)
void prewitt_kernel(const float* __restrict__ x,
                    float* __restrict__ mag_out,
                    float* __restrict__ mask_out) {
    __shared__ float smem[PIX_PER_BLOCK];

    const unsigned tid       = threadIdx.x;
    const unsigned blockBase = blockIdx.x * (unsigned)PIX_PER_BLOCK;

    // ---- Stage 8 images into LDS: exactly 7 async b128 copies per thread --
    const float* gbase = x + blockBase;
#pragma unroll
    for (int k = 0; k < VEC4_PER_BLOCK / THREADS; ++k) {
        const unsigned i = tid + (unsigned)k * THREADS;
        __builtin_amdgcn_global_load_async_to_lds_b128(
            (AS1 v4i*)(gbase + 4u * i),
            (AS3 v4i*)(smem  + 4u * i),
            /*offset=*/0, /*cpol=*/0);
    }
    __builtin_amdgcn_s_wait_asynccnt(0);
    __syncthreads();

    // ---- One thread handles one full image row ----------------------------
    const unsigned img = tid / W;           // 0..7   (computed once)
    const unsigned r   = tid - img * W;     // 0..27

    const float* im   = smem + img * IMG_PIX;
    const float* rowM = im + r * W;
    const float* rowT = (r > 0)     ? rowM - W : rowM;   // clamped (valid LDS)
    const float* rowB = (r < W - 1) ? rowM + W : rowM;
    const float selT  = (r > 0)     ? 1.f : 0.f;
    const float selB  = (r < W - 1) ? 1.f : 0.f;

    // Load the 3 rows (21 ds_load_b128), form D = bot - top immediately.
    float M[W];   // middle row
    float D[W];   // selB*B - selT*T
#pragma unroll
    for (int k = 0; k < W / 4; ++k) {
        const v4f mm = ((const v4f*)rowM)[k];
        const v4f tt = ((const v4f*)rowT)[k];
        const v4f bb = ((const v4f*)rowB)[k];
        const v4f dd = bb * selB - tt * selT;
        M[4 * k + 0] = mm.x; M[4 * k + 1] = mm.y; M[4 * k + 2] = mm.z; M[4 * k + 3] = mm.w;
        D[4 * k + 0] = dd.x; D[4 * k + 1] = dd.y; D[4 * k + 2] = dd.z; D[4 * k + 3] = dd.w;
    }

    // Vertical 3-tap via shared partial sums: P[j] = D[j] + D[j+1].
    float P[W - 1];
#pragma unroll
    for (int j = 0; j < W - 1; ++j) P[j] = D[j] + D[j + 1];

    float mag[W];
#pragma unroll
    for (int j = 0; j < W; ++j) {
        // v_j = D[j-1] + D[j] + D[j+1], zero-padded at the ends
        const float v = (j == 0)     ? P[0]
                      : (j == W - 1) ? P[W - 2]
                                     : P[j - 1] + D[j + 1];
        // h_j = M[j+1] - M[j-1], zero-padded
        const float h = (j == 0)     ? M[1]
                      : (j == W - 1) ? -M[W - 2]
                                     : M[j + 1] - M[j - 1];
        mag[j] = __builtin_amdgcn_sqrtf(v * v + h * h + 1e-6f);
    }

    // ---- Stream out: 7 + 7 b128 NT stores per row --------------------------
    const unsigned gi = blockBase + tid * W;   // img*784 + r*28
#pragma unroll
    for (int k = 0; k < W / 4; ++k) {
        v4f mg, mk;
        mg.x = mag[4 * k + 0]; mg.y = mag[4 * k + 1];
        mg.z = mag[4 * k + 2]; mg.w = mag[4 * k + 3];
        mk.x = mg.x > 0.f ? 1.f : 0.f;
        mk.y = mg.y > 0.f ? 1.f : 0.f;
        mk.z = mg.z > 0.f ? 1.f : 0.f;
        mk.w = mg.w > 0.f ? 1.f : 0.f;
        __builtin_nontemporal_store(mg, (v4f*)(mag_out  + gi + 4u * k));
        __builtin_nontemporal_store(mk, (v4f*)(mask_out + gi + 4u * k));
    }
}

extern "C" void kernel_launch(void* const* d_in, const int* in_sizes, int n_in,
                              void* d_out, int out_size, void* d_ws, size_t ws_size,
                              hipStream_t stream) {
    const float* x   = (const float*)d_in[0];
    float*       out = (float*)d_out;

    const long long total = (long long)in_sizes[0];       // B * 784
    const int       B     = (int)(total / IMG_PIX);       // 65536

    float* mag  = out;
    float* mask = out + total;   // tuple outputs concatenated flat

    const int blocks = (B + IMGS_PER_BLOCK - 1) / IMGS_PER_BLOCK; // 8192
    prewitt_kernel<<<blocks, THREADS, 0, stream>>>(x, mag, mask);
}